// FractalDim3d_28209345200701
// MI455X (gfx1250) — compile-verified
//
#include <hip/hip_runtime.h>
#include <hip/hip_bf16.h>
#include <stdint.h>

// FractalDim3d for MI455X (gfx1250):
//  kernel 1: per-32^3-block pyramid moments, TDM tensor_load_to_lds tile fetch
//  kernel 2: deterministic block-sum + least-squares slope via v_wmma_f32_16x16x4_f32

typedef float v2f  __attribute__((ext_vector_type(2)));
typedef float v8f  __attribute__((ext_vector_type(8)));
typedef unsigned int u32x4 __attribute__((ext_vector_type(4)));
typedef int  i32x8 __attribute__((ext_vector_type(8)));
typedef int  i32x4 __attribute__((ext_vector_type(4)));

#define NK 8
#define IMG 128
#define BLK 32
#define NBC 6
#define NBLK 64        // 4x4x4 blocks per channel image
#define WS_STRIDE 64   // floats per (bc,blk) record: 48 moments + 1 block sum

// accumulate 8 moment slots for one (nonzero-masked) value
__device__ __forceinline__ void accum8(float* a, float x, const float* kf, const bool* isone) {
  if (x != 0.0f) {
    float lx  = __logf(x);
    float xlx = x * lx;
#pragma unroll
    for (int k = 0; k < NK; ++k)
      a[k] += isone[k] ? xlx : __expf(kf[k] * lx);
  }
}

__global__ __launch_bounds__(256) void fd_moments_kernel(const float* __restrict__ img,
                                                         const int* __restrict__ bounds,
                                                         float* __restrict__ ws) {
  __shared__ float tile[BLK * BLK * BLK]; // 128 KB: level 0 tile
  __shared__ float l1[16 * 16 * 16];
  __shared__ float l2[8 * 8 * 8];
  __shared__ float l3[4 * 4 * 4];
  __shared__ float l4[8];
  __shared__ float accW[8][48];

  const int tid = threadIdx.x;
  const int bc  = blockIdx.x >> 6;
  const int blk = blockIdx.x & 63;
  const int bx = blk & 3, by = (blk >> 2) & 3, bz = blk >> 4;
  const long long base = (long long)bc * (IMG * IMG * IMG)
                       + (long long)(bx * BLK)
                       + (long long)(by * BLK) * IMG
                       + (long long)(bz * BLK) * IMG * IMG;

  float kf[NK]; bool isone[NK];
#pragma unroll
  for (int k = 0; k < NK; ++k) { int b = bounds[k]; kf[k] = (float)b; isone[k] = (b == 1); }

  // ---- fetch 32^3 f32 tile into LDS ----
#if defined(__gfx1250__) && __has_builtin(__builtin_amdgcn_tensor_load_to_lds)
  if (tid < 32) { // wave 0 issues the TDM op (EXEC-independent, one per wave)
    unsigned long long ga = (unsigned long long)(uintptr_t)(img + base);
    typedef __attribute__((address_space(3))) char lds_char;
    unsigned lds_off = (unsigned)(unsigned long long)(lds_char*)&tile[0];
    // D# group0: count=1 | lds_addr | global_addr[56:0] | type=2
    u32x4 g0 = { 1u,
                 lds_off,
                 (unsigned)(ga & 0xffffffffull),
                 (unsigned)((ga >> 32) & 0x01ffffffull) | 0x80000000u };
    // D# group1: data_size=4B; tensor 128x128(x128); tile 32x32x32; strides 128, 16384
    i32x8 g1 = { 0x00020000,   // workgroup_mask=0, data_size=2 (4B), no flags
                 0x00800000,   // tensor_dim0 = 128 (bits 63:48)
                 0x00800000,   // tensor_dim1 = 128 (bits 95:80)
                 0x00200000,   // tile_dim0   = 32  (bits 127:112)
                 0x00200020,   // tile_dim1=32, tile_dim2=32
                 128,          // tensor_dim0_stride = 128
                 0x40000000,   // tensor_dim1_stride = 16384 (bits 223:208)
                 0 };
    // D# group2: tensor_dim2=128, tensor_dim3=0, tensor_dim2_stride=2097152, tile_dim3=0
    i32x4 g2 = { 128, 0, 0x00200000, 0 };
    i32x4 g3 = { 0, 0, 0, 0 };
    // clang-23 (amdgpu-toolchain) takes a 6-arg form with an extra i32x8 group; zero-fill it
    i32x8 g4 = { 0, 0, 0, 0, 0, 0, 0, 0 };
    __builtin_amdgcn_tensor_load_to_lds(g0, g1, g2, g3, g4, 0);
#if __has_builtin(__builtin_amdgcn_s_wait_tensorcnt)
    __builtin_amdgcn_s_wait_tensorcnt(0);
#else
    asm volatile("s_wait_tensorcnt 0" ::: "memory");
#endif
  }
#else
  for (int i = tid; i < BLK * BLK * BLK; i += 256) {
    int x = i & 31, y = (i >> 5) & 31, z = i >> 10;
    tile[i] = img[base + x + (long long)y * IMG + (long long)z * IMG * IMG];
  }
#endif
  __syncthreads();

  float acc[48];
#pragma unroll
  for (int j = 0; j < 48; ++j) acc[j] = 0.0f;

  // level 0: 32^3
  for (int i = tid; i < 32768; i += 256) accum8(acc + 0, tile[i], kf, isone);

  // level 1: 16^3
  for (int c = tid; c < 4096; c += 256) {
    int x = c & 15, y = (c >> 4) & 15, z = c >> 8;
    int b0 = 2 * x + 2 * y * 32 + 2 * z * 1024;
    float s = tile[b0] + tile[b0 + 1] + tile[b0 + 32] + tile[b0 + 33]
            + tile[b0 + 1024] + tile[b0 + 1025] + tile[b0 + 1056] + tile[b0 + 1057];
    l1[c] = s;
    accum8(acc + 8, s, kf, isone);
  }
  __syncthreads();

  // level 2: 8^3
  if (tid < 512) {
    int c = tid;
    int x = c & 7, y = (c >> 3) & 7, z = c >> 6;
    int b0 = 2 * x + 2 * y * 16 + 2 * z * 256;
    float s = l1[b0] + l1[b0 + 1] + l1[b0 + 16] + l1[b0 + 17]
            + l1[b0 + 256] + l1[b0 + 257] + l1[b0 + 272] + l1[b0 + 273];
    l2[c] = s;
    accum8(acc + 16, s, kf, isone);
  }
  __syncthreads();

  // level 3: 4^3
  if (tid < 64) {
    int c = tid;
    int x = c & 3, y = (c >> 2) & 3, z = c >> 4;
    int b0 = 2 * x + 2 * y * 8 + 2 * z * 64;
    float s = l2[b0] + l2[b0 + 1] + l2[b0 + 8] + l2[b0 + 9]
            + l2[b0 + 64] + l2[b0 + 65] + l2[b0 + 72] + l2[b0 + 73];
    l3[c] = s;
    accum8(acc + 24, s, kf, isone);
  }
  __syncthreads();

  // level 4: 2^3
  if (tid < 8) {
    int c = tid;
    int x = c & 1, y = (c >> 1) & 1, z = c >> 2;
    int b0 = 2 * x + 2 * y * 4 + 2 * z * 16;
    float s = l3[b0] + l3[b0 + 1] + l3[b0 + 4] + l3[b0 + 5]
            + l3[b0 + 16] + l3[b0 + 17] + l3[b0 + 20] + l3[b0 + 21];
    l4[c] = s;
    accum8(acc + 32, s, kf, isone);
  }
  __syncthreads();

  // level 5: 1 cell == whole block sum
  float bsum = 0.0f;
  if (tid == 0) {
    float s = l4[0] + l4[1] + l4[2] + l4[3] + l4[4] + l4[5] + l4[6] + l4[7];
    bsum = s;
    accum8(acc + 40, s, kf, isone);
  }

  // deterministic reduction: wave shuffles, then cross-wave via LDS
  const int lane = tid & 31, wv = tid >> 5;
#pragma unroll
  for (int j = 0; j < 48; ++j) {
    float v = acc[j];
#pragma unroll
    for (int o = 16; o > 0; o >>= 1) v += __shfl_down(v, o, 32);
    if (lane == 0) accW[wv][j] = v;
  }
  __syncthreads();
  if (tid < 48) {
    float r = 0.0f;
#pragma unroll
    for (int w = 0; w < 8; ++w) r += accW[w][tid];
    ws[(size_t)blockIdx.x * WS_STRIDE + tid] = r;
  }
  if (tid == 0) ws[(size_t)blockIdx.x * WS_STRIDE + 48] = bsum;
}

// p(bc,k,s) from per-channel sums held in LDS
__device__ __forceinline__ float fd_p(const float* S, const int* bounds, int bc, int k, int s) {
  float b  = S[bc * 49 + 48];
  float lb = __logf(b);
  int   kb = bounds[k];
  float M  = S[bc * 49 + s * 8 + k];
  return (kb == 1) ? (M / b - lb) : (__logf(M) - (float)kb * lb);
}

__global__ __launch_bounds__(32) void fd_finalize_kernel(const float* __restrict__ ws,
                                                         const int* __restrict__ bounds,
                                                         float* __restrict__ out) {
  __shared__ float S[NBC * 49];
  __shared__ float dtile[16];
  const int tid = threadIdx.x;

  // deterministic block-sum of the 384 partial records
  for (int i = tid; i < NBC * 49; i += 32) {
    int bc = i / 49, slot = i - bc * 49;
    float v = 0.0f;
    for (int b = 0; b < NBLK; ++b)
      v += ws[(size_t)(bc * NBLK + b) * WS_STRIDE + slot];
    S[i] = v;
  }
  __syncthreads();

  // out[r] = a_k * sum_s p(s)*(6s-15) / (105*ln2)
  const float inv_den = 1.0f / (105.0f * 0.69314718055994530942f);
  for (int t = 0; t < 3; ++t) {
    const int m  = tid & 15;
    const int kp = (tid >> 4) << 1;          // K-pair offset for this lane half
    const int r  = t * 16 + m;
    const int bc = r >> 3, k = r & 7;
#if __has_builtin(__builtin_amdgcn_wmma_f32_16x16x4_f32)
    // A[m][s] = p(s) * (6s-15); B = all ones -> D[m][*] = row sums (weight folded into A)
    float a0x = fd_p(S, bounds, bc, k, kp)     * (float)(6 * kp - 15);
    float a0y = fd_p(S, bounds, bc, k, kp + 1) * (float)(6 * (kp + 1) - 15);
    int s2 = kp + 4, s3 = kp + 5;
    float a1x = (s2 < 6) ? fd_p(S, bounds, bc, k, s2) * (float)(6 * s2 - 15) : 0.0f;
    float a1y = (s3 < 6) ? fd_p(S, bounds, bc, k, s3) * (float)(6 * s3 - 15) : 0.0f;
    v2f A0 = { a0x, a0y };
    v2f A1 = { a1x, a1y };
    v2f B1 = { 1.0f, 1.0f };
    v8f c  = {};
    c = __builtin_amdgcn_wmma_f32_16x16x4_f32(false, A0, false, B1, (short)0, c, false, false);
    c = __builtin_amdgcn_wmma_f32_16x16x4_f32(false, A1, false, B1, (short)0, c, false, false);
    // C/D layout: lane 0 vgpr j = D[j][0] (rows 0..7); lane 16 vgpr j = D[j+8][0]
    if (tid == 0) {
#pragma unroll
      for (int j = 0; j < 8; ++j) dtile[j] = c[j];
    }
    if (tid == 16) {
#pragma unroll
      for (int j = 0; j < 8; ++j) dtile[8 + j] = c[j];
    }
#else
    if (tid < 16) {
      float srow = 0.0f;
      for (int s = 0; s < 6; ++s) srow += fd_p(S, bounds, bc, k, s) * (float)(6 * s - 15);
      dtile[tid] = srow;
    }
#endif
    __syncthreads();
    if (tid < 16) {
      int rr = t * 16 + tid, kk = rr & 7, kb = bounds[kk];
      float a = (kb == 1) ? 1.0f : 1.0f / ((float)kb - 1.0f);
      out[rr] = a * dtile[tid] * inv_den;
    }
    __syncthreads();
  }
}

extern "C" void kernel_launch(void* const* d_in, const int* in_sizes, int n_in,
                              void* d_out, int out_size, void* d_ws, size_t ws_size,
                              hipStream_t stream) {
  const float* img    = (const float*)d_in[0];
  const int*   bounds = (const int*)d_in[1];
  float*       ws     = (float*)d_ws;
  float*       out    = (float*)d_out;
  (void)in_sizes; (void)n_in; (void)out_size; (void)ws_size;

  fd_moments_kernel<<<dim3(NBC * NBLK), dim3(256), 0, stream>>>(img, bounds, ws);
  fd_finalize_kernel<<<dim3(1), dim3(32), 0, stream>>>(ws, bounds, out);
}